// BilateralSlice_51092930953397
// MI455X (gfx1250) — compile-verified
//
#include <hip/hip_runtime.h>

// -----------------------------------------------------------------------------
// HDRNet bilateral_slice on MI455X (gfx1250, wave32).
//
// Roofline: out 201MB + guide 16MB + grid 0.4MB => ~9.4us @ 23.3 TB/s; the
// kernel is instruction-issue-limited, so the design minimizes instr/tile.
//
//  * Each 16-pixel row tile: D[16x16] = A[16x32 weights] x B[32x16 grid
//    corners] via 8 chained V_WMMA_F32_16X16X4_F32 (K = 2x*2y*8z candidate
//    corners). Full fp32, bit-faithful to the reference weight math.
//  * Block = 4 rows x 2 x-tiles = 8 tiles sharing one B matrix (fx uniform on
//    32-aligned spans, fy uniform on 4-aligned row quads; 1/64-cell
//    breakpoints at 64n+32 respect both alignments). B's 16 loads +
//    zero-pad selects and all cell scalars amortize 8x.
//  * All 8 guide values prefetched at block top -> no mid-block load stalls.
//  * z-mass closed form: zm(k)=max(0,1-|k+0.5-clamp(8*guide,0.5,7.5)|); each
//    lane needs only 4 distinct k across all K-slices -> 12 VALU/tile.
//  * D repacked through a wave-private LDS slice -> 3 coalesced 256B stores.
// -----------------------------------------------------------------------------

typedef float v2f __attribute__((ext_vector_type(2)));
typedef float v8f __attribute__((ext_vector_type(8)));

#define NB 4
#define GH 16
#define GW 16
#define GD 8
#define GC 12
#define IH 1024
#define IW 1024
// Block = 4 rows x 32 px (2 WMMA tiles) = 8 tiles.
#define BLK_X (IW / 32)                  // 32
#define BLK_Y (IH / 4)                   // 256
#define N_BLOCKS (NB * BLK_Y * BLK_X)    // 32768

__global__ __launch_bounds__(256) void bslice_wmma_kernel(
    const float* __restrict__ grid,
    const float* __restrict__ guide,
    float* __restrict__ out)
{
    __shared__ float lds[8 * 192];       // 192 floats (16px x 12ch) per wave
    const int lane = threadIdx.x & 31;
    const int wave = threadIdx.x >> 5;
    float* tlds = lds + wave * 192;

    const int   hi   = lane >> 4;        // half-wave: K offset +2 for A and B
    const int   p    = lane & 15;        // pixel row M (A,D) == column N (B,D)
    const int   cmin = min(p, GC - 1);   // clamped B column address
    const bool  cok  = (p < GC);         // zero-pad channels 12..15
    const float pf   = (float)p;
    const float k05  = 2.0f * (float)hi + 0.5f;  // lane's base z sample point

    const int waves_total = gridDim.x * 8;
    int wid = blockIdx.x * 8 + wave;

    for (int b = wid; b < N_BLOCKS; b += waves_total) {
        const int n   = b >> 13;                 // / (256*32)
        const int rem = b & 8191;
        const int y0  = (rem >> 5) << 2;         // 4-aligned row quad
        const int x0  = (rem & 31) << 5;         // 32-aligned, 2 x-tiles

        // ---- prefetch all 8 guide values (pixel p of each tile) ----
        const float* gptr = guide + (n * IH + y0) * IW + x0 + p;
        float gv[4][2];
        #pragma unroll
        for (int r = 0; r < 4; ++r)
            #pragma unroll
            for (int t = 0; t < 2; ++t)
                gv[r][t] = gptr[r * IW + t * 16];

        // ---- y cell (uniform over the row quad) ----
        const float fyf = floorf(((float)y0 + 0.5f) * (1.0f / 64.0f) - 0.5f);
        const int   fy  = (int)fyf;
        const int   cy0 = min(max(fy, 0), GH - 1);
        const int   cy1 = min(max(fy + 1, 0), GH - 1);

        // ---- x cell (uniform over the 32px span) ----
        const float fxf = floorf(((float)x0 + 0.5f) * (1.0f / 64.0f) - 0.5f);
        const int   fx  = (int)fxf;
        const int   cx0 = min(max(fx, 0), GW - 1);
        const int   cx1 = min(max(fx + 1, 0), GW - 1);

        // per-pixel x weights for both x-tiles (pure f32, exact dyadics)
        float wx1t[2], wx0t[2];
        #pragma unroll
        for (int t = 0; t < 2; ++t) {
            const float gx = ((float)(x0 + 16 * t) + pf + 0.5f) * (1.0f / 64.0f);
            wx1t[t] = gx - 0.5f - fxf;
            wx0t[t] = 1.0f - wx1t[t];
        }

        // per-row y weights
        float wy0r[4], wy1r[4];
        #pragma unroll
        for (int r = 0; r < 4; ++r) {
            const float gy = ((float)(y0 + r) + 0.5f) * (1.0f / 64.0f);
            wy1r[r] = gy - 0.5f - fyf;
            wy0r[r] = 1.0f - wy1r[r];
        }

        // ---- B matrix: shared by all 8 tiles of the block ----
        // corner idx = i*16 + j*8 + k; slice s covers K=4s..4s+3:
        //   i=s>>2, j=(s>>1)&1, k=(s&1)*4 + hi*2 + {0,1}
        const float* gb[2][2];
        gb[0][0] = grid + ((((n * GH + cy0) * GW + cx0) * GD) * GC);
        gb[0][1] = grid + ((((n * GH + cy1) * GW + cx0) * GD) * GC);
        gb[1][0] = grid + ((((n * GH + cy0) * GW + cx1) * GD) * GC);
        gb[1][1] = grid + ((((n * GH + cy1) * GW + cx1) * GD) * GC);

        v2f Bs[8];
        #pragma unroll
        for (int s = 0; s < 8; ++s) {
            const int i  = s >> 2;
            const int j  = (s >> 1) & 1;
            const int ke = (s & 1) * 4 + hi * 2;
            const float* base = gb[i][j];
            const float b0 = base[ke * GC + cmin];
            const float b1 = base[(ke + 1) * GC + cmin];
            Bs[s].x = cok ? b0 : 0.0f;
            Bs[s].y = cok ? b1 : 0.0f;
        }

        #pragma unroll
        for (int r = 0; r < 4; ++r) {
            const float wy0 = wy0r[r];
            const float wy1 = wy1r[r];
            float* orow = out + ((size_t)(n * IH + (y0 + r)) * IW + x0) * GC;

            #pragma unroll
            for (int t = 0; t < 2; ++t) {
                // ---- per-pixel z masses: closed-form clipped hat ----
                const float gz  = gv[r][t] * (float)GD;
                const float gzc = fminf(fmaxf(gz, 0.5f), 7.5f);
                float zm[4];    // k = 2hi+{0,1}, 2hi+4+{0,1}
                zm[0] = fmaxf(1.0f - fabsf(k05        - gzc), 0.0f);
                zm[1] = fmaxf(1.0f - fabsf(k05 + 1.0f - gzc), 0.0f);
                zm[2] = fmaxf(1.0f - fabsf(k05 + 4.0f - gzc), 0.0f);
                zm[3] = fmaxf(1.0f - fabsf(k05 + 5.0f - gzc), 0.0f);

                const float wxy[2][2] = {
                    { wx0t[t] * wy0, wx0t[t] * wy1 },
                    { wx1t[t] * wy0, wx1t[t] * wy1 } };

                v8f acc = {};
                #pragma unroll
                for (int s = 0; s < 8; ++s) {
                    const int   i   = s >> 2;
                    const int   j   = (s >> 1) & 1;
                    const int   zi  = (s & 1) * 2;
                    const float wij = wxy[i][j];
                    v2f a;
                    a.x = wij * zm[zi];
                    a.y = wij * zm[zi + 1];
                    acc = __builtin_amdgcn_wmma_f32_16x16x4_f32(
                        false, a, false, Bs[s], (short)0, acc, false, false);
                }

                // ---- repack D via wave-private LDS -> coalesced stores ----
                // D layout: VGPR q -> pixel M = q + 8*hi, channel N = p.
                if (cok) {
                    #pragma unroll
                    for (int q = 0; q < 8; ++q) {
                        const int m = q + hi * 8;
                        tlds[m * GC + p] = acc[q];
                    }
                }
                float* obase = orow + t * (16 * GC);    // 192 contiguous floats
                #pragma unroll
                for (int q = 0; q < 3; ++q) {
                    const int e = q * 64 + lane * 2;
                    const float2 v = *(const float2*)(tlds + e);
                    *(float2*)(obase + e) = v;          // 256B contiguous/store
                }
            }
        }
    }
}

extern "C" void kernel_launch(void* const* d_in, const int* in_sizes, int n_in,
                              void* d_out, int out_size, void* d_ws, size_t ws_size,
                              hipStream_t stream) {
    (void)in_sizes; (void)n_in; (void)out_size; (void)d_ws; (void)ws_size;
    const float* grid  = (const float*)d_in[0];
    const float* guide = (const float*)d_in[1];
    float*       out   = (float*)d_out;
    // 1024 blocks x 8 waves = 8192 waves; 4 blocks each = 32768 blocks exactly.
    bslice_wmma_kernel<<<1024, 256, 0, stream>>>(grid, guide, out);
}